// TorchKitNET_41412074668784
// MI455X (gfx1250) — compile-verified
//
#include <hip/hip_runtime.h>
#include <hip/hip_bf16.h>

typedef float v2f __attribute__((ext_vector_type(2)));
typedef float v8f __attribute__((ext_vector_type(8)));

#define NB 262144
#define NC 16
#define NF 16
#define NH 12

constexpr int XS = 260;            // padded x-tile row stride (dwords); 260 % 64 == 4 -> conflict-free strided reads
constexpr int HS = 20;             // h-buffer row stride (dwords)
constexpr int WAVES_PER_BLOCK = 2;
constexpr int TILES_PER_WAVE = 8;
constexpr int NBLOCKS = NB / (16 * TILES_PER_WAVE * WAVES_PER_BLOCK); // 1024
constexpr int NWAVES  = NBLOCKS * WAVES_PER_BLOCK;                    // 2048

// sigmoid with hardware transcendentals: v_exp_f32 + v_rcp_f32 (no IEEE divide path)
__device__ __forceinline__ float sigmoidf_fast(float x) {
    const float e = __builtin_amdgcn_exp2f(x * -1.442695040888963f); // exp(-x)
    return __builtin_amdgcn_rcpf(1.0f + e);
}

__global__ __launch_bounds__(WAVES_PER_BLOCK * 32)
void kitnet_main(const float* __restrict__ x,
                 const float* __restrict__ We,   // [C][H][F]
                 const float* __restrict__ be,   // [C][H]
                 const float* __restrict__ Wd,   // [C][F][H]
                 const float* __restrict__ bd,   // [C][F]
                 const int*   __restrict__ cidx, // [C][F]
                 float* __restrict__ ws_part)    // [NWAVES][C]
{
    __shared__ float xs[WAVES_PER_BLOCK][16 * XS];
    __shared__ float hs[WAVES_PER_BLOCK][16 * HS];
    __shared__ int   idx_s[NC * NF];

    const int lane  = threadIdx.x & 31;
    const int wid   = threadIdx.x >> 5;
    const int gwave = blockIdx.x * WAVES_PER_BLOCK + wid;
    const int half  = lane >> 4;   // 0: K lo pair, 1: K hi pair
    const int l16   = lane & 15;   // M row (A) / N col (B, C, D)

    for (int i = threadIdx.x; i < NC * NF; i += WAVES_PER_BLOCK * 32)
        idx_s[i] = cidx[i];
    __syncthreads();

    float acc[NC];
#pragma unroll
    for (int c = 0; c < NC; ++c) acc[c] = 0.0f;

    float* myx = xs[wid];
    float* myh = hs[wid];

    for (int t = 0; t < TILES_PER_WAVE; ++t) {
        const int row0 = (gwave * TILES_PER_WAVE + t) * 16;

        // ---- stage 16 rows x 256 cols, fully coalesced ----
#pragma unroll 4
        for (int r = 0; r < 16; ++r) {
            const float4* src = (const float4*)(x + (size_t)(row0 + r) * 256);
            float4 v0 = src[lane];
            float4 v1 = src[lane + 32];
            *(float4*)(myx + r * XS + lane * 4)       = v0;
            *(float4*)(myx + r * XS + 128 + lane * 4) = v1;
            if (t + 1 < TILES_PER_WAVE)  // warm L2/WGP$ for next tile (global_prefetch_b8)
                __builtin_prefetch(x + (size_t)(row0 + 16 + r) * 256 + lane * 8, 0, 0);
        }
        // wave-private LDS region + per-wave in-order DS: no barrier needed

        for (int c = 0; c < NC; ++c) {
            // ---- encode: D = xc(16x16) * We_c^T(16x12pad16), K split into 4x WMMA f32 16x16x4 ----
            v8f d = {0.f, 0.f, 0.f, 0.f, 0.f, 0.f, 0.f, 0.f};
#pragma unroll
            for (int s = 0; s < 4; ++s) {
                const int k0 = 4 * s + half * 2;
                const int i0 = idx_s[c * NF + k0];      // broadcast LDS read
                const int i1 = idx_s[c * NF + k0 + 1];
                v2f a;
                a.x = myx[l16 * XS + i0];               // A: lane=row, vgprs=K pair
                a.y = myx[l16 * XS + i1];
                v2f b; b.x = 0.0f; b.y = 0.0f;
                if (l16 < NH) {                          // pad N=12..15 with zeros
                    const float* wp = We + (size_t)(c * NH + l16) * NF + k0;
                    b.x = wp[0];
                    b.y = wp[1];
                }
                d = __builtin_amdgcn_wmma_f32_16x16x4_f32(
                        false, a, false, b, (short)0, d, false, false);
            }

            // ---- bias + sigmoid; write h (D layout) to LDS for layout transpose ----
            const float bev = (l16 < NH) ? be[c * NH + l16] : 0.0f;
#pragma unroll
            for (int j = 0; j < 8; ++j) {
                const float hv = sigmoidf_fast(d[j] + bev);
                myh[(j + half * 8) * HS + l16] = hv;     // row M = j + half*8, col N = l16
            }

            // ---- decode: D2 = h(16x12) * Wd_c^T(12x16), 3x WMMA f32 16x16x4 ----
            v8f d2 = {0.f, 0.f, 0.f, 0.f, 0.f, 0.f, 0.f, 0.f};
#pragma unroll
            for (int s = 0; s < 3; ++s) {
                const int k0 = 4 * s + half * 2;         // K in [0,12)
                v2f a;
                a.x = myh[l16 * HS + k0];                // h re-read in A layout
                a.y = myh[l16 * HS + k0 + 1];
                const float* wp = Wd + (size_t)(c * NF + l16) * NH + k0;
                v2f b; b.x = wp[0]; b.y = wp[1];
                d2 = __builtin_amdgcn_wmma_f32_16x16x4_f32(
                        false, a, false, b, (short)0, d2, false, false);
            }

            // ---- rec = sigmoid(D2 + bd); accumulate squared reconstruction error ----
            const float bdv = bd[c * NF + l16];
            const int   col = idx_s[c * NF + l16];
            float part = 0.0f;
#pragma unroll
            for (int j = 0; j < 8; ++j) {
                const float rec = sigmoidf_fast(d2[j] + bdv);
                const float xv  = myx[(j + half * 8) * XS + col];
                const float e   = rec - xv;
                part += e * e;
            }
            acc[c] += part;
        }
    }

    // ---- wave reduction -> per-wave partial SSE per cluster (deterministic, no atomics) ----
#pragma unroll
    for (int c = 0; c < NC; ++c) {
        float v = acc[c];
#pragma unroll
        for (int off = 16; off > 0; off >>= 1)
            v += __shfl_xor(v, off, 32);
        if (lane == 0)
            ws_part[gwave * NC + c] = v;
    }
}

__global__ __launch_bounds__(256)
void kitnet_head(const float* __restrict__ ws_part, // [NWAVES][C]
                 const float* __restrict__ He,      // [HH=12][C=16]
                 const float* __restrict__ hbe,     // [12]
                 const float* __restrict__ Hd,      // [C=16][HH=12]
                 const float* __restrict__ hbd,     // [16]
                 float* __restrict__ out)           // [head_out(16), tails(16)]
{
    __shared__ float red[NC][17];
    __shared__ float tails[NC];
    __shared__ float h2[12];

    const int t = threadIdx.x;
    const int c = t & 15;
    const int chunk = t >> 4;            // 16 chunks of NWAVES/16 each
    const int per = NWAVES / 16;         // 128

    float s = 0.0f;
    for (int i = chunk * per; i < (chunk + 1) * per; ++i)
        s += ws_part[i * NC + c];
    red[c][chunk] = s;
    __syncthreads();

    if (t < NC) {
        float tot = 0.0f;
#pragma unroll
        for (int k = 0; k < 16; ++k) tot += red[t][k];
        float loss = sqrtf(tot / (float)((double)NB * NF));
        if (loss == 0.0f) loss = 0.01f;  // jnp.where(loss==0, 0.01, loss)
        tails[t] = loss;                 // /(1+1e-16) is exact identity in fp32
    }
    __syncthreads();

    if (t < 12) {
        float a = hbe[t];
#pragma unroll
        for (int k = 0; k < NC; ++k) a += He[t * NC + k] * tails[k];
        h2[t] = sigmoidf_fast(a);
    }
    __syncthreads();

    if (t < NC) {
        float a = hbd[t];
#pragma unroll
        for (int k = 0; k < 12; ++k) a += Hd[t * 12 + k] * h2[k];
        out[t]      = sigmoidf_fast(a);  // head_out
        out[NC + t] = tails[t];          // tails
    }
}

extern "C" void kernel_launch(void* const* d_in, const int* in_sizes, int n_in,
                              void* d_out, int out_size, void* d_ws, size_t ws_size,
                              hipStream_t stream) {
    const float* x    = (const float*)d_in[0];
    const float* We   = (const float*)d_in[1];
    const float* be   = (const float*)d_in[2];
    const float* Wd   = (const float*)d_in[3];
    const float* bd   = (const float*)d_in[4];
    const float* He   = (const float*)d_in[5];
    const float* hbe  = (const float*)d_in[6];
    const float* Hd   = (const float*)d_in[7];
    const float* hbd  = (const float*)d_in[8];
    const int*   cidx = (const int*)d_in[9];

    float* part = (float*)d_ws;          // NWAVES*NC floats = 128 KB

    kitnet_main<<<NBLOCKS, WAVES_PER_BLOCK * 32, 0, stream>>>(
        x, We, be, Wd, bd, cidx, part);
    kitnet_head<<<1, 256, 0, stream>>>(
        part, He, hbe, Hd, hbd, (float*)d_out);
}